// ResGraphModule_76785425318277
// MI455X (gfx1250) — compile-verified
//
#include <hip/hip_runtime.h>

#define N_NODES 100000
#define N_EDGES 600000
#define DIM     128
#define EPS_F   1e-5f

// Padded LDS row stride (floats): keeps float4 stores 16B-aligned
// (132 % 4 == 0) and spreads ds_load_b64 fragment reads across banks
// (132 % 64 == 4 -> lane m hits banks {4m,4m+1} / {4m+2,4m+3}).
#define LDS_STRIDE 132

typedef __attribute__((ext_vector_type(2))) float v2f;
typedef __attribute__((ext_vector_type(8))) float v8f;

// ---------------------------------------------------------------------------
// Kernel 1: zero the aggregation buffer (d_out doubles as aggr accumulator).
// N*DIM = 12.8M floats, divisible by 4 -> float4 stores.
// ---------------------------------------------------------------------------
__global__ __launch_bounds__(256) void zero_f4(float* __restrict__ p, int n4) {
  int i = blockIdx.x * blockDim.x + threadIdx.x;
  if (i < n4) {
    float4 z = {0.f, 0.f, 0.f, 0.f};
    ((float4*)p)[i] = z;
  }
}

// ---------------------------------------------------------------------------
// Kernel 2: edge phase. One thread per (edge, 4-feature chunk):
//   msg = relu(x[src] + edge_attr);  atomicAdd(aggr[dst], msg)
// 600000 * 32 = 19.2M threads. Atomics resolve in L2 (192MB >> 51MB aggr).
// ---------------------------------------------------------------------------
__global__ __launch_bounds__(256) void edge_scatter(
    const float*     __restrict__ x,
    const long long* __restrict__ ei,   // [2, E] int64
    const float*     __restrict__ ea,   // [E, D]
    float*           __restrict__ aggr) // [N, D]
{
  long long idx = (long long)blockIdx.x * blockDim.x + threadIdx.x;
  int e = (int)(idx >> 5);        // DIM/4 = 32 chunks per edge
  if (e >= N_EDGES) return;
  int c = (int)(idx & 31) << 2;

  long long s = ei[e];
  long long d = ei[(long long)N_EDGES + e];

  float4 xv = *(const float4*)(x  + s * DIM + c);
  float4 av = *(const float4*)(ea + (long long)e * DIM + c);

  float m0 = fmaxf(xv.x + av.x, 0.f);
  float m1 = fmaxf(xv.y + av.y, 0.f);
  float m2 = fmaxf(xv.z + av.z, 0.f);
  float m3 = fmaxf(xv.w + av.w, 0.f);

  float* a = aggr + d * DIM + c;
  atomicAdd(a + 0, m0);
  atomicAdd(a + 1, m1);
  atomicAdd(a + 2, m2);
  atomicAdd(a + 3, m3);
}

// ---------------------------------------------------------------------------
// Kernel 3: fused GEMM + epilogue using V_WMMA_F32_16X16X4_F32.
//   A = (1+eps)*x + aggr           [16 x 128]  (staged in LDS, shared by 8 waves)
//   B = W^T                        [128 x 128] (B[k][n] = W[n][k])
//   out = relu(A@B + bias) + x
// Block: 256 thr = 8 waves; wave w owns output columns [16w, 16w+16).
// Grid: N/16 = 6250 blocks (exact).
//
// Fragment layouts per CDNA5 ISA (16x4 f32 A, mirrored B):
//   A: lane L -> row m = L&15, VGPR0/1 = K-pair at koff = (L>>4)*2
//   B: lane L -> col n = L&15, VGPR0/1 = W[n0+n][k0+koff], [k0+koff+1]
//   C: VGPR p -> row p + (L>=16 ? 8 : 0), col = L&15
// Reads aggr/x rows [16*bid, 16*bid+16) fully before writing the same rows:
// aggr aliases `out` safely (block-private rows, __syncthreads barrier).
// ---------------------------------------------------------------------------
__global__ __launch_bounds__(256) void gine_gemm_epilogue(
    const float* __restrict__ x,
    const float* __restrict__ aggr,
    const float* __restrict__ W,     // [D, D] row-major
    const float* __restrict__ bias,  // [D]
    float*       __restrict__ out)   // [N, D]
{
  __shared__ float As[16 * LDS_STRIDE];

  const int node0 = blockIdx.x * 16;
  const int tid   = threadIdx.x;

  // --- Stage A-tile: 2048 floats / 256 threads = 8 floats (2 x float4) each.
  {
    int base = tid * 8;
    int row  = base >> 7;        // /128
    int col  = base & 127;
    long long g = (long long)(node0 + row) * DIM + col;
    #pragma unroll
    for (int u = 0; u < 8; u += 4) {
      float4 xv = *(const float4*)(x    + g + u);
      float4 av = *(const float4*)(aggr + g + u);
      float4 t;
      t.x = (1.0f + EPS_F) * xv.x + av.x;
      t.y = (1.0f + EPS_F) * xv.y + av.y;
      t.z = (1.0f + EPS_F) * xv.z + av.z;
      t.w = (1.0f + EPS_F) * xv.w + av.w;
      *(float4*)(&As[row * LDS_STRIDE + col + u]) = t;
    }
  }
  __syncthreads();

  const int wave = tid >> 5;
  const int lane = tid & 31;
  const int n0   = wave * 16;
  const int mrow = lane & 15;         // A row / B col within tile
  const int koff = (lane >> 4) * 2;   // K-pair select per half-wave

  const float* arow = As + mrow * LDS_STRIDE + koff;
  const float* wrow = W + (n0 + mrow) * DIM + koff;

  v8f c = {};
  #pragma unroll
  for (int k0 = 0; k0 < DIM; k0 += 4) {
    v2f af = *(const v2f*)(arow + k0);   // ds_load_b64
    v2f bf = *(const v2f*)(wrow + k0);   // global_load_b64
    c = __builtin_amdgcn_wmma_f32_16x16x4_f32(
        /*neg_a=*/false, af, /*neg_b=*/false, bf,
        /*c_mod=*/(short)0, c, /*reuse_a=*/false, /*reuse_b=*/false);
  }

  // --- Epilogue: bias, relu, residual. C layout: VGPR p -> (m, n) below.
  const int col   = n0 + (lane & 15);
  const int mbase = (lane >> 4) * 8;
  const float bcol = bias[col];
  #pragma unroll
  for (int p = 0; p < 8; ++p) {
    int m = mbase + p;
    long long g = (long long)(node0 + m) * DIM + col;
    float h = c[p] + bcol;
    out[g] = fmaxf(h, 0.f) + x[g];
  }
}

// ---------------------------------------------------------------------------
extern "C" void kernel_launch(void* const* d_in, const int* in_sizes, int n_in,
                              void* d_out, int out_size, void* d_ws, size_t ws_size,
                              hipStream_t stream) {
  const float*     x  = (const float*)d_in[0];
  const long long* ei = (const long long*)d_in[1];
  const float*     ea = (const float*)d_in[2];
  const float*     W  = (const float*)d_in[3];
  const float*     b  = (const float*)d_in[4];
  float* out = (float*)d_out;   // also serves as the aggr accumulator

  (void)in_sizes; (void)n_in; (void)d_ws; (void)ws_size; (void)out_size;

  // 1) aggr = 0   (N*DIM = 12,800,000 floats -> 3.2M float4)
  {
    int n4 = (N_NODES * DIM) / 4;
    int blocks = (n4 + 255) / 256;
    zero_f4<<<blocks, 256, 0, stream>>>(out, n4);
  }
  // 2) scatter-add relu(x[src] + edge_attr) into aggr (== out)
  {
    long long total = (long long)N_EDGES * (DIM / 4);
    int blocks = (int)((total + 255) / 256);
    edge_scatter<<<blocks, 256, 0, stream>>>(x, ei, ea, out);
  }
  // 3) fused GEMM + epilogue (reads its own 16 rows of out-as-aggr, then writes)
  {
    int blocks = N_NODES / 16;   // 6250, exact
    gine_gemm_epilogue<<<blocks, 256, 0, stream>>>(x, out, W, b, out);
  }
}